// SS2D_2156073582803
// MI455X (gfx1250) — compile-verified
//
#include <hip/hip_runtime.h>

typedef float v2f __attribute__((ext_vector_type(2)));
typedef float v8f __attribute__((ext_vector_type(8)));

namespace {
constexpr int B_ = 2, H_ = 48, W_ = 48, DM = 96, DE = 192, K_ = 4, N_ = 16, R_ = 6;
constexpr int L_ = H_ * W_;      // 2304
constexpr int CT = R_ + 2 * N_;  // 38 (dt rows 0..5, B rows 6..21, C rows 22..37)
constexpr int CP = 48;           // padded to 3 WMMA tiles
constexpr int BL = B_ * L_;      // 4608
}

__device__ __forceinline__ float silu_f(float x) { return x / (1.0f + __expf(-x)); }

// map scan position m of direction k -> row-major spatial index h*W+w
__device__ __forceinline__ int spmap(int k, int m) {
  if (k & 1) m = L_ - 1 - m;                       // reversed scans
  if (k >= 2) {                                    // vertical scans: m = w*H + h
    int w = m / H_;
    int h = m - w * H_;
    m = h * W_ + w;
  }
  return m;
}

// ---------------------------------------------------------------------------
// K1: xz = x @ in_proj_w^T ; split into xh and silu(z).  f32 WMMA 16x16x4.
// ---------------------------------------------------------------------------
__global__ void k_in_proj(const float* __restrict__ x, const float* __restrict__ w,
                          float* __restrict__ xh, float* __restrict__ zs) {
  const int mt = blockIdx.x * 16;      // rows of x (b*L + l)
  const int nt = blockIdx.y * 16;      // output channel tile (0..383)
  const int lane = threadIdx.x & 31;
  const int lid = lane & 15, half = lane >> 4;

  const float* ap = x + (size_t)(mt + lid) * DM + half * 2;  // A: lane=M, half picks K pair
  const float* bp = w + (size_t)(nt + lid) * DM + half * 2;  // B[kr,n] = W[n, kr]
  v8f acc = {};
  for (int kk = 0; kk < DM; kk += 4) {
    v2f a, b;
    a.x = ap[kk]; a.y = ap[kk + 1];
    b.x = bp[kk]; b.y = bp[kk + 1];
    acc = __builtin_amdgcn_wmma_f32_16x16x4_f32(false, a, false, b, (short)0, acc, false, false);
  }
  const int d = nt + lid;
#pragma unroll
  for (int r = 0; r < 8; ++r) {
    const int row = mt + r + half * 8;
    const float v = acc[r];
    if (d < DE) xh[(size_t)row * DE + d] = v;
    else        zs[(size_t)row * DE + (d - DE)] = silu_f(v);
  }
}

// ---------------------------------------------------------------------------
// K2: depthwise 3x3 SAME conv + bias + silu (channel-last).
// ---------------------------------------------------------------------------
__global__ void k_dwconv(const float* __restrict__ xh, const float* __restrict__ cw,
                         const float* __restrict__ cb, float* __restrict__ xc) {
  const int idx = blockIdx.x * blockDim.x + threadIdx.x;
  if (idx >= BL * DE) return;
  const int d = idx % DE;
  const int sp = idx / DE;
  const int w = sp % W_;
  const int h = (sp / W_) % H_;
  const int b = sp / (W_ * H_);
  const float* wp = cw + d * 9;
  float acc = cb[d];
#pragma unroll
  for (int i = 0; i < 3; ++i) {
    const int hh = h + i - 1;
    if (hh < 0 || hh >= H_) continue;
#pragma unroll
    for (int j = 0; j < 3; ++j) {
      const int ww = w + j - 1;
      if (ww < 0 || ww >= W_) continue;
      acc += wp[i * 3 + j] * xh[((size_t)(b * H_ + hh) * W_ + ww) * DE + d];
    }
  }
  xc[idx] = silu_f(acc);
}

// ---------------------------------------------------------------------------
// K3: x_dbl[b,k,l,c] = sum_d xs[b,k,d,l] * xpw[k,c,d]   (stored transposed, CP=48)
//     xs gathered from xc via spmap — no materialization of the 4 orderings.
// ---------------------------------------------------------------------------
__global__ void k_xdbl(const float* __restrict__ xc, const float* __restrict__ xpw,
                       float* __restrict__ xdbl) {
  const int gt = blockIdx.x * 16;  // global columns g = b*L + l
  const int ct = blockIdx.y * 16;  // c tile (0,16,32)
  const int k  = blockIdx.z;
  const int lane = threadIdx.x & 31;
  const int lid = lane & 15, half = lane >> 4;
  const int b = gt / L_;
  const int l0 = gt - b * L_;

  const int c = ct + lid;
  const bool cv = (c < CT);
  const float* ap = xpw + ((size_t)k * CT + (cv ? c : 0)) * DE + half * 2;
  const int sp = spmap(k, l0 + lid);
  const float* bp = xc + ((size_t)b * L_ + sp) * DE + half * 2;

  v8f acc = {};
  for (int kk = 0; kk < DE; kk += 4) {
    v2f a, bv;
    a.x = cv ? ap[kk] : 0.0f;
    a.y = cv ? ap[kk + 1] : 0.0f;
    bv.x = bp[kk]; bv.y = bp[kk + 1];
    acc = __builtin_amdgcn_wmma_f32_16x16x4_f32(false, a, false, bv, (short)0, acc, false, false);
  }
  // D tile: row c = ct + r + 8*half, col l = l0 + lid; transposed store -> 8 contiguous floats
  float* orow = xdbl + (((size_t)(b * K_ + k) * L_) + (l0 + lid)) * CP + ct + half * 8;
#pragma unroll
  for (int r = 0; r < 8; ++r) orow[r] = acc[r];
}

// ---------------------------------------------------------------------------
// K4: sequential selective scan. One wave = 2 (b,k,d) channels; lane&15 = state n.
//     Fuses dt-projection + softplus + recurrence + C-contraction.
// ---------------------------------------------------------------------------
__global__ void k_scan(const float* __restrict__ xc, const float* __restrict__ xdbl,
                       const float* __restrict__ dtw, const float* __restrict__ dtb,
                       const float* __restrict__ alogs, const float* __restrict__ dsv,
                       float* __restrict__ ys) {
  const int lane = threadIdx.x & 31;
  const int lid = lane & 15, half = lane >> 4;
  const int ch = blockIdx.x * 2 + half;  // channel = ((b*K + k)*DE + d)
  const int d = ch % DE;
  const int k = (ch / DE) % K_;
  const int b = ch / (DE * K_);

  const float An = -__expf(alogs[((size_t)(k * DE + d)) * N_ + lid]);
  const float Dv = dsv[k * DE + d];
  const float bias = dtb[k * DE + d];
  float wr[R_];
#pragma unroll
  for (int r = 0; r < R_; ++r) wr[r] = dtw[((size_t)(k * DE + d)) * R_ + r];

  const float* rowbase = xdbl + (size_t)(b * K_ + k) * L_ * CP;
  const float* xcb = xc + (size_t)b * L_ * DE + d;
  float* yout = ys + (size_t)ch * L_;

  float h = 0.0f;
  for (int l = 0; l < L_; ++l) {
    const float* row = rowbase + (size_t)l * CP;
    __builtin_prefetch(row + CP, 0, 0);  // global_prefetch next descriptor row
    float draw = bias;
#pragma unroll
    for (int r = 0; r < R_; ++r) draw += wr[r] * row[r];
    const float delta = (draw > 20.0f) ? draw : log1pf(__expf(draw));
    const float u = xcb[(size_t)spmap(k, l) * DE];
    const float Bn = row[R_ + lid];
    const float Cn = row[R_ + N_ + lid];
    h = h * __expf(delta * An) + (delta * u) * Bn;
    float t = h * Cn;
    t += __shfl_xor(t, 8);  // half-wave (16-lane) reduction: stays within each channel
    t += __shfl_xor(t, 4);
    t += __shfl_xor(t, 2);
    t += __shfl_xor(t, 1);
    if (lid == 0) yout[l] = t + Dv * u;
  }
}

// ---------------------------------------------------------------------------
// K5: combine 4 directions (gather), LayerNorm over DE, multiply by silu(z).
//     One wave per spatial position (6 channels per lane).
// ---------------------------------------------------------------------------
__global__ void k_combine_ln(const float* __restrict__ ys, const float* __restrict__ zs,
                             const float* __restrict__ gamma, const float* __restrict__ beta,
                             float* __restrict__ yz) {
  const int wid = threadIdx.x >> 5;
  const int lane = threadIdx.x & 31;
  const int g = blockIdx.x * (blockDim.x >> 5) + wid;
  if (g >= BL) return;
  const int b = g / L_;
  const int sp = g - b * L_;
  const int w = sp % W_;
  const int h = sp / W_;
  const int m2 = w * H_ + h;  // vertical-scan position that lands on (h,w)

  float v[6];
  float s = 0.0f, s2 = 0.0f;
#pragma unroll
  for (int i = 0; i < 6; ++i) {
    const int d = lane + 32 * i;
    const float t = ys[(((size_t)(b * K_ + 0)) * DE + d) * L_ + sp]
                  + ys[(((size_t)(b * K_ + 1)) * DE + d) * L_ + (L_ - 1 - sp)]
                  + ys[(((size_t)(b * K_ + 2)) * DE + d) * L_ + m2]
                  + ys[(((size_t)(b * K_ + 3)) * DE + d) * L_ + (L_ - 1 - m2)];
    v[i] = t; s += t; s2 += t * t;
  }
#pragma unroll
  for (int off = 16; off >= 1; off >>= 1) {
    s  += __shfl_xor(s, off);
    s2 += __shfl_xor(s2, off);
  }
  const float mean = s * (1.0f / DE);
  const float var = s2 * (1.0f / DE) - mean * mean;
  const float rstd = rsqrtf(var + 1e-5f);
#pragma unroll
  for (int i = 0; i < 6; ++i) {
    const int d = lane + 32 * i;
    const float ln = (v[i] - mean) * rstd * gamma[d] + beta[d];
    yz[(size_t)g * DE + d] = ln * zs[(size_t)g * DE + d];
  }
}

// ---------------------------------------------------------------------------
// K6: out = yz @ out_proj_w^T   (4608x192 x 192x96), f32 WMMA.
// ---------------------------------------------------------------------------
__global__ void k_out_proj(const float* __restrict__ yz, const float* __restrict__ ow,
                           float* __restrict__ out) {
  const int mt = blockIdx.x * 16;
  const int nt = blockIdx.y * 16;
  const int lane = threadIdx.x & 31;
  const int lid = lane & 15, half = lane >> 4;
  const float* ap = yz + (size_t)(mt + lid) * DE + half * 2;
  const float* bp = ow + (size_t)(nt + lid) * DE + half * 2;
  v8f acc = {};
  for (int kk = 0; kk < DE; kk += 4) {
    v2f a, b;
    a.x = ap[kk]; a.y = ap[kk + 1];
    b.x = bp[kk]; b.y = bp[kk + 1];
    acc = __builtin_amdgcn_wmma_f32_16x16x4_f32(false, a, false, b, (short)0, acc, false, false);
  }
#pragma unroll
  for (int r = 0; r < 8; ++r) {
    const int row = mt + r + half * 8;
    out[(size_t)row * DM + nt + lid] = acc[r];
  }
}

extern "C" void kernel_launch(void* const* d_in, const int* in_sizes, int n_in,
                              void* d_out, int out_size, void* d_ws, size_t ws_size,
                              hipStream_t stream) {
  (void)in_sizes; (void)n_in; (void)out_size; (void)ws_size;
  const float* x   = (const float*)d_in[0];
  const float* ipw = (const float*)d_in[1];
  const float* cw  = (const float*)d_in[2];
  const float* cb  = (const float*)d_in[3];
  const float* xpw = (const float*)d_in[4];
  const float* dtw = (const float*)d_in[5];
  const float* dtb = (const float*)d_in[6];
  const float* alg = (const float*)d_in[7];
  const float* dsv = (const float*)d_in[8];
  const float* gam = (const float*)d_in[9];
  const float* bet = (const float*)d_in[10];
  const float* opw = (const float*)d_in[11];
  float* out = (float*)d_out;

  float* ws = (float*)d_ws;
  float* xh   = ws;                                   // BL*DE
  float* zs   = xh + (size_t)BL * DE;                 // BL*DE
  float* xc   = zs + (size_t)BL * DE;                 // BL*DE
  float* xdbl = xc + (size_t)BL * DE;                 // B*K*L*CP
  float* ys   = xdbl + (size_t)B_ * K_ * L_ * CP;     // B*K*DE*L
  float* yz   = ys + (size_t)B_ * K_ * DE * L_;       // BL*DE

  k_in_proj<<<dim3(BL / 16, (2 * DE) / 16), 32, 0, stream>>>(x, ipw, xh, zs);
  k_dwconv<<<(BL * DE + 255) / 256, 256, 0, stream>>>(xh, cw, cb, xc);
  k_xdbl<<<dim3(BL / 16, CP / 16, K_), 32, 0, stream>>>(xc, xpw, xdbl);
  k_scan<<<(B_ * K_ * DE) / 2, 32, 0, stream>>>(xc, xdbl, dtw, dtb, alg, dsv, ys);
  k_combine_ln<<<(BL + 7) / 8, 256, 0, stream>>>(ys, zs, gam, bet, yz);
  k_out_proj<<<dim3(BL / 16, DM / 16), 32, 0, stream>>>(yz, opw, out);
}